// IoU3DLossVariablePointHead_85650237817254
// MI455X (gfx1250) — compile-verified
//
#include <hip/hip_runtime.h>
#include <math.h>

#define EPSF   1e-6f
#define CLIPV  1.0e7f
#define NPTS   24
#define NPAD   32
#define FMAXV  3.4028234663852886e38f

// jnp.nan_to_num defaults: nan->0, +inf->FLT_MAX, -inf->-FLT_MAX
__device__ __forceinline__ float nan0(float x) {
  if (__builtin_isnan(x)) return 0.0f;
  if (x ==  INFINITY) return  FMAXV;
  if (x == -INFINITY) return -FMAXV;
  return x;
}

__device__ __forceinline__ float clipc(float x) {
  return fminf(fmaxf(x, -CLIPV), CLIPV);
}

// Strictly monotone surrogate of atan2(y,x): maps angle (-pi,pi] -> (-2,2].
// The angles feed ONLY argsort, so the resulting order (hence the output)
// is identical while saving ~24 transcendental atan2 calls per box.
__device__ __forceinline__ float pseudo_angle(float y, float x) {
  float d = fmaxf(fabsf(x) + fabsf(y), 1e-38f);
  float t = y / d;
  return (x >= 0.f) ? t : ((y >= 0.f) ? (2.f - t) : (-2.f - t));
}

// Branchless compare-exchange carrying (key, x, y). The point's mask is
// encoded in the key (active keys in [-2,2], inactive in [998,1002]) and
// recovered after the sort, so it does not ride through the network.
// Select-based key swap (not min/max) keeps key/payload consistent even
// for NaN keys in degenerate configurations.
__device__ __forceinline__ void ce3(float& ka, float& kb, float& xa, float& xb,
                                    float& ya, float& yb) {
  const bool s = ka > kb;
  float t;
  t = s ? kb : ka; kb = s ? ka : kb; ka = t;
  t = s ? xb : xa; xb = s ? xa : xb; xa = t;
  t = s ? yb : ya; yb = s ? ya : yb; ya = t;
}

__global__ __launch_bounds__(256) void iou3d_vph_kernel(
    const float* __restrict__ base_coors,   // N x 3
    const float* __restrict__ pred_logits,  // N x 8 (16B-aligned rows)
    const float* __restrict__ gt_attrs,     // N x 7
    const float* __restrict__ anchor,       // 3
    float* __restrict__ out, int n)
{
  const int i = blockIdx.x * blockDim.x + threadIdx.x;
  if (i >= n) return;

  // ---- loads (compiler clause-groups these as b96/b128) ----
  const float4* lg = (const float4*)(pred_logits + (size_t)i * 8);
  const float4 L0 = lg[0];
  const float4 L1 = lg[1];
  const float* bc = base_coors + (size_t)i * 3;
  const float bx0 = bc[0], bx1 = bc[1], bx2 = bc[2];
  const float* ga = gt_attrs + (size_t)i * 7;
  const float gw = ga[0], gl = ga[1], gh = ga[2];
  const float gx = ga[3], gy = ga[4], gz = ga[5], gr = ga[6];
  const float a0 = anchor[0], a1 = anchor[1], a2 = anchor[2];

  // ---- _roi_logits_to_attrs ----
  const float diag = sqrtf(a0 * a0 + a1 * a1);
  const float px = clipc(L0.x * diag + bx0);
  const float py = clipc(L0.y * diag + bx1);
  const float pz = clipc(L0.z * diag + bx2);
  const float pw = fminf(fmaxf(expf(L0.w) * a0, 0.f), CLIPV);
  const float pl = fminf(fmaxf(expf(L1.x) * a1, 0.f), CLIPV);
  const float ph = fminf(fmaxf(expf(L1.y) * a2, 0.f), CLIPV);
  const float pr = clipc(atan2f(L1.z, L1.w));

  // ---- rotations / offsets ----
  // |gr|,|pr| <= pi and |relr| <= 2pi by construction, so the hardware
  // v_sin_f32/v_cos_f32 fast path (valid to +-512pi) is exact-range here;
  // __sincosf avoids dragging in the Payne-Hanek slow path 3x.
  const float relr = pr - gr;
  float crr, srr, cg, sg, cp, sp;
  __sincosf(relr, &srr, &crr);
  __sincosf(gr,   &sg,  &cg);
  __sincosf(pr,   &sp,  &cp);
  const float rx = px - gx, ry = py - gy;
  // Rot(-gt_r) @ rel_xy
  const float opx =  cg * rx + sg * ry;
  const float opy = -sg * rx + cg * ry;
  // Rot(-pred_r) @ (-rel_xy)
  const float ogx = -(cp * rx + sp * ry);
  const float ogy =   sp * rx - cp * ry;

  const float hwp = 0.5f * pw, hlp = 0.5f * pl;
  const float hwg = 0.5f * gw, hlg = 0.5f * gl;

  // box vertex pattern: v0=(hw,-hl) v1=(hw,hl) v2=(-hw,hl) v3=(-hw,-hl)
  const float sxv[4] = { 1.f, 1.f, -1.f, -1.f };
  const float syv[4] = { -1.f, 1.f, 1.f, -1.f };

  float ptx[NPTS], pty[NPTS], msk[NPTS];
  float Px[4], Py[4];

  // pred vertices rotated into gt frame; gt vertices rotated into pred frame
  #pragma unroll
  for (int k = 0; k < 4; ++k) {
    const float vx = sxv[k] * hwp, vy = syv[k] * hlp;   // pred_v
    Px[k] = crr * vx - srr * vy + opx;                  // rel_rot_pred_v
    Py[k] = srr * vx + crr * vy + opy;
    const float gvx = sxv[k] * hwg, gvy = syv[k] * hlg; // gt_v
    const float Gx =  crr * gvx + srr * gvy + ogx;      // rel_rot_gt_v
    const float Gy = -srr * gvx + crr * gvy + ogy;
    // pts[0..3] = gt_v, mask = gt_in_pred (slack 0)
    ptx[k] = gvx; pty[k] = gvy;
    msk[k] = (fabsf(Gx) <= hwp && fabsf(Gy) <= hlp) ? 1.f : 0.f;
    // pts[4..7] = rel_rot_pred_v, mask = pred_in_gt (slack 0)
    ptx[4 + k] = Px[k]; pty[4 + k] = Py[k];
    msk[4 + k] = (fabsf(Px[k]) <= hwg && fabsf(Py[k]) <= hlg) ? 1.f : 0.f;
  }

  // ---- _intersection_points: edges (v[e-1], v[e]) vs gt-aligned lines ----
  #pragma unroll
  for (int e = 0; e < 4; ++e) {
    const float v0x = Px[(e + 3) & 3], v0y = Py[(e + 3) & 3];
    const float v1x = Px[e],           v1y = Py[e];
    const float dx = v1x - v0x + EPSF;
    const float dy = v1y - v0y + EPSF;
    const float kx  = nan0((v1y - v0y) / dx);
    const float bxi = nan0((v0y * v1x - v1y * v0x) / dx);
    const float ky  = nan0((v1x - v0x) / dy);
    const float byi = nan0((v1y * v0x - v0y * v1x) / dy);
    const int b = 8 + e * 4;
    ptx[b + 0] =  hwg;             pty[b + 0] =  kx * hwg + bxi;
    ptx[b + 1] = -hwg;             pty[b + 1] = -kx * hwg + bxi;
    ptx[b + 2] =  ky * hlg + byi;  pty[b + 2] =  hlg;
    ptx[b + 3] = -ky * hlg + byi;  pty[b + 3] = -hlg;
  }

  // inter masks: inside gt (slack 1e-3) AND inside pred (slack 1e-3, pred frame)
  #pragma unroll
  for (int j = 8; j < NPTS; ++j) {
    const bool mg = (fabsf(ptx[j]) <= hwg + 1e-3f) && (fabsf(pty[j]) <= hlg + 1e-3f);
    const float qx =  crr * ptx[j] + srr * pty[j] + ogx;   // Rot(-rel_r)p + offG
    const float qy = -srr * ptx[j] + crr * pty[j] + ogy;
    const bool mp = (fabsf(qx) <= hwp + 1e-3f) && (fabsf(qy) <= hlp + 1e-3f);
    msk[j] = (mg && mp) ? 1.f : 0.f;
  }

  // ---- centroid of masked points (0/0 -> 0 per nan_to_num) ----
  float smm = 0.f, sx = 0.f, sy = 0.f;
  #pragma unroll
  for (int k = 0; k < NPTS; ++k) {
    smm += msk[k];
    sx += ptx[k] * msk[k];
    sy += pty[k] * msk[k];
  }
  const float cx = nan0(sx / smm);
  const float cy = nan0(sy / smm);

  // ---- sort keys: surrogate angle + 1000*(1-mask). msk[] dies here; the
  // mask is recovered from the sorted key (key < 500 <=> mask == 1).
  const float bxr = ptx[0] - cx, byr = pty[0] - cy;
  float key[NPTS];
  #pragma unroll
  for (int k = 0; k < NPTS; ++k) {
    const float rxk = ptx[k] - cx, ryk = pty[k] - cy;
    const float dot = bxr * rxk + byr * ryk;
    const float det = bxr * ryk - byr * rxk;
    key[k] = pseudo_angle(det + EPSF, dot + EPSF) + (1.f - msk[k]) * 1000.f;
  }

  // ---- Batcher odd-even mergesort, n=32 network pruned to 24 real slots
  // (valid by the +inf-padding argument: any comparator touching a virtual
  // slot >= 24 is a no-op). All indices are compile-time after unrolling,
  // so the arrays stay entirely in VGPRs.
  #pragma unroll
  for (int p = 1; p < NPAD; p <<= 1) {
    #pragma unroll
    for (int q = p; q >= 1; q >>= 1) {
      #pragma unroll
      for (int j = q & (p - 1); j + q < NPAD; j += 2 * q) {
        #pragma unroll
        for (int t = 0; t < q; ++t) {
          const int lo = j + t, hi = j + t + q;
          if (hi < NPTS && (lo / (2 * p)) == (hi / (2 * p)))
            ce3(key[lo], key[hi], ptx[lo], ptx[hi], pty[lo], pty[hi]);
        }
      }
    }
  }

  // ---- recover mask from key; clip + mask in place (sp = clip(sorted)*sm) ----
  #pragma unroll
  for (int k = 0; k < NPTS; ++k) {
    const float m = (key[k] < 500.f) ? 1.f : 0.f;
    ptx[k] = clipc(ptx[k]) * m;
    pty[k] = clipc(pty[k]) * m;
  }

  // last active vertex (Python-style modulo on -1 -> 23)
  int last_id = (int)(smm + 0.5f) - 1;
  if (last_id < 0) last_id += NPTS;
  float lvx = 0.f, lvy = 0.f;
  #pragma unroll
  for (int k = 0; k < NPTS; ++k) {
    const bool h = (k == last_id);
    lvx = h ? ptx[k] : lvx;
    lvy = h ? pty[k] : lvy;
  }

  // ---- shoelace over padded polygon [last, p0..p23] ----
  float area = lvx * pty[0] - ptx[0] * lvy;
  #pragma unroll
  for (int k = 0; k < NPTS - 1; ++k)
    area += ptx[k] * pty[k + 1] - ptx[k + 1] * pty[k];
  area *= 0.5f;

  // ---- vertical overlap + IoU ----
  const float top = fminf(gz + 0.5f * gh, pz + 0.5f * ph);
  const float bot = fmaxf(gz - 0.5f * gh, pz - 0.5f * ph);
  const float ihh = fmaxf(top - bot, 0.f);
  const float iv  = area * ihh;
  const float gvol = gw * gl * gh;
  const float pvol = pw * pl * ph;
  out[i] = nan0(iv / (gvol + pvol - iv));
}

extern "C" void kernel_launch(void* const* d_in, const int* in_sizes, int n_in,
                              void* d_out, int out_size, void* d_ws, size_t ws_size,
                              hipStream_t stream) {
  (void)n_in; (void)d_ws; (void)ws_size; (void)out_size;
  const float* base_coors  = (const float*)d_in[0];
  const float* pred_logits = (const float*)d_in[1];
  const float* gt_attrs    = (const float*)d_in[2];
  const float* anchor      = (const float*)d_in[3];
  float* out = (float*)d_out;

  const int n = in_sizes[0] / 3;  // base_coors is N x 3
  const int block = 256;          // 8 wave32s
  const int grid = (n + block - 1) / block;
  hipLaunchKernelGGL(iou3d_vph_kernel, dim3(grid), dim3(block), 0, stream,
                     base_coors, pred_logits, gt_attrs, anchor, out, n);
}